// GraphClassifier_33964601377212
// MI455X (gfx1250) — compile-verified
//
#include <hip/hip_runtime.h>
#include <hip/hip_bf16.h>
#include <math.h>

typedef float v2f __attribute__((ext_vector_type(2)));
typedef float v8f __attribute__((ext_vector_type(8)));

#define IN_F 64
#define HID 64
#define OUT_F 10

static __device__ __forceinline__ void atomic_fadd(float* p, float v) {
    __hip_atomic_fetch_add(p, v, __ATOMIC_RELAXED, __HIP_MEMORY_SCOPE_AGENT);
}

// ---------------------------------------------------------------- utilities
__global__ void fill_kernel(float* __restrict__ p, float v, int n) {
    int i = blockIdx.x * blockDim.x + threadIdx.x;
    if (i < n) p[i] = v;
}

__global__ void deg_kernel(const int* __restrict__ dst, float* __restrict__ deg, int n_edges) {
    int e = blockIdx.x * blockDim.x + threadIdx.x;
    if (e < n_edges) atomic_fadd(&deg[dst[e]], 1.0f);
}

__global__ void dinv_kernel(const float* __restrict__ deg, float* __restrict__ dinv, int n) {
    int i = blockIdx.x * blockDim.x + threadIdx.x;
    if (i < n) dinv[i] = rsqrtf(fmaxf(deg[i], 1.0f));
}

// ---------------------------------------------------------------- WMMA GEMM
// hs[row, :] = (X[row, :] @ W) * dinv[row]     X:[M,64] W:[64,64] row-major
// One wave32 per 16x16 output tile; K=64 via 16 x V_WMMA_F32_16X16X4_F32.
__global__ void gemm64_scale_wmma(const float* __restrict__ X,
                                  const float* __restrict__ W,
                                  const float* __restrict__ dinv,
                                  float* __restrict__ out,
                                  int m_tiles) {
    int wave = (blockIdx.x * blockDim.x + threadIdx.x) >> 5;
    int lane = threadIdx.x & 31;
    int n_tile = wave & 3;          // 64 cols -> 4 N-tiles
    int m_tile = wave >> 2;
    if (m_tile >= m_tiles) return;  // wave-uniform guard: EXEC stays all-1s

    const int row0 = m_tile * 16;
    const int n0   = n_tile * 16;
    const int half = lane >> 4;     // 0: lanes 0-15, 1: lanes 16-31
    const int l16  = lane & 15;

    // A: lane l<16 holds (M=l, K=k0..k0+1); lane l>=16 holds (M=l-16, K=k0+2..k0+3)
    const float* xrow = X + (size_t)(row0 + l16) * IN_F + 2 * half;

    v8f acc = {0.f, 0.f, 0.f, 0.f, 0.f, 0.f, 0.f, 0.f};
#pragma unroll
    for (int k0 = 0; k0 < IN_F; k0 += 4) {
        v2f a;
        a.x = xrow[k0 + 0];
        a.y = xrow[k0 + 1];
        // B: VGPR v holds K = k0 + v + 2*half, N = n0 + l16
        int kb = k0 + 2 * half;
        v2f b;
        b.x = W[(size_t)(kb + 0) * HID + n0 + l16];
        b.y = W[(size_t)(kb + 1) * HID + n0 + l16];
        acc = __builtin_amdgcn_wmma_f32_16x16x4_f32(
            /*neg_a=*/false, a, /*neg_b=*/false, b,
            /*c_mod=*/(short)0, acc, /*reuse_a=*/false, /*reuse_b=*/false);
    }

    // C/D layout: VGPR v -> M = v + 8*half, N = n0 + l16. Fuse row scaling.
#pragma unroll
    for (int v = 0; v < 8; ++v) {
        int row = row0 + v + 8 * half;
        out[(size_t)row * HID + n0 + l16] = acc[v] * dinv[row];
    }
}

// ---------------------------------------------------------------- aggregation
// agg[dst] += hs[src]  (64 floats per edge; 32 lanes/edge, float2 per lane)
__global__ void edge_agg_kernel(const float* __restrict__ hs,
                                const int* __restrict__ src,
                                const int* __restrict__ dst,
                                float* __restrict__ agg,
                                int n_edges) {
    int t = blockIdx.x * blockDim.x + threadIdx.x;
    int e = t >> 5;
    int lane = t & 31;
    if (e >= n_edges) return;
    int s = src[e];
    int d = dst[e];
    float2 v = ((const float2*)(hs + (size_t)s * HID))[lane];
    float* dp = agg + (size_t)d * HID + 2 * lane;
    atomic_fadd(dp + 0, v.x);
    atomic_fadd(dp + 1, v.y);
}

// out = relu((agg + hs_selfloop) * dinv[node] + bias)
__global__ void finalize_relu_kernel(const float* __restrict__ agg,
                                     const float* __restrict__ hs,
                                     const float* __restrict__ dinv,
                                     const float* __restrict__ bias,
                                     float* __restrict__ out,
                                     int n_nodes) {
    int t = blockIdx.x * blockDim.x + threadIdx.x;
    int node = t >> 6;
    int f = t & 63;
    if (node >= n_nodes) return;
    float v = (agg[t] + hs[t]) * dinv[node] + bias[f];
    out[t] = v > 0.f ? v : 0.f;
}

// ---------------------------------------------------------------- pooling
__global__ void pool_kernel(const float* __restrict__ h,
                            const int* __restrict__ batch,
                            float* __restrict__ pooled,
                            float* __restrict__ cnts,
                            int n_nodes) {
    int t = blockIdx.x * blockDim.x + threadIdx.x;
    int node = t >> 6;
    int f = t & 63;
    if (node >= n_nodes) return;
    int g = batch[node];
    atomic_fadd(&pooled[(size_t)g * HID + f], h[t]);
    if (f == 0) atomic_fadd(&cnts[g], 1.0f);
}

// ---------------------------------------------------------------- classifier
// one block (64 threads) per graph: mean -> logits(10) -> log_softmax
__global__ void classify_kernel(const float* __restrict__ pooled,
                                const float* __restrict__ cnts,
                                const float* __restrict__ Wc,   // [64,10]
                                const float* __restrict__ bc,   // [10]
                                float* __restrict__ out) {      // [G,10]
    __shared__ float mean_s[HID];
    __shared__ float logit_s[OUT_F];
    int g = blockIdx.x;
    int t = threadIdx.x;
    float c = cnts[g];
    c = c < 1.0f ? 1.0f : c;
    mean_s[t] = pooled[(size_t)g * HID + t] / c;
    __syncthreads();
    if (t < OUT_F) {
        float acc = bc[t];
#pragma unroll
        for (int k = 0; k < HID; ++k) acc += mean_s[k] * Wc[k * OUT_F + t];
        logit_s[t] = acc;
    }
    __syncthreads();
    if (t == 0) {
        float mx = logit_s[0];
#pragma unroll
        for (int j = 1; j < OUT_F; ++j) mx = fmaxf(mx, logit_s[j]);
        float s = 0.f;
#pragma unroll
        for (int j = 0; j < OUT_F; ++j) s += __expf(logit_s[j] - mx);
        float lse = mx + __logf(s);
#pragma unroll
        for (int j = 0; j < OUT_F; ++j) out[(size_t)g * OUT_F + j] = logit_s[j] - lse;
    }
}

// ---------------------------------------------------------------- launcher
extern "C" void kernel_launch(void* const* d_in, const int* in_sizes, int n_in,
                              void* d_out, int out_size, void* d_ws, size_t ws_size,
                              hipStream_t stream) {
    const float* x     = (const float*)d_in[0];   // [N,64]
    const int*   eidx  = (const int*)d_in[1];     // [2,E]
    const int*   batch = (const int*)d_in[2];     // [N]
    const float* W1    = (const float*)d_in[3];
    const float* b1    = (const float*)d_in[4];
    const float* W2    = (const float*)d_in[5];
    const float* b2    = (const float*)d_in[6];
    const float* Wc    = (const float*)d_in[7];
    const float* bc    = (const float*)d_in[8];
    float* out = (float*)d_out;

    const int n_nodes  = in_sizes[0] / IN_F;      // 50000
    const int n_edges  = in_sizes[1] / 2;         // 800000
    const int n_graphs = out_size / OUT_F;        // 500
    const int* src = eidx;
    const int* dst = eidx + n_edges;

    // workspace layout (floats)
    float* ws     = (float*)d_ws;
    float* deg    = ws;                               // [N]
    float* dinv   = deg + n_nodes;                    // [N]
    float* hs     = dinv + n_nodes;                   // [N,64] scaled gemm out
    float* agg    = hs + (size_t)n_nodes * HID;       // [N,64]
    float* hbuf   = agg + (size_t)n_nodes * HID;      // [N,64] layer output
    float* pooled = hbuf + (size_t)n_nodes * HID;     // [G,64]
    float* cnts   = pooled + (size_t)n_graphs * HID;  // [G]

    const int B = 256;
    const int nfeat   = n_nodes * HID;
    const int m_tiles = (n_nodes + 15) / 16;
    const int gwaves  = m_tiles * 4;                  // one wave per 16x16 tile
    const int gblocks = (gwaves + 7) / 8;             // 8 waves per 256-thr block
    const int eblocks = (int)(((size_t)n_edges * 32 + B - 1) / B);
    const int fblocks = (nfeat + B - 1) / B;

    // degrees (self-loop folded in as deg=1 init) and dinv
    fill_kernel<<<(n_nodes + B - 1) / B, B, 0, stream>>>(deg, 1.0f, n_nodes);
    deg_kernel<<<(n_edges + B - 1) / B, B, 0, stream>>>(dst, deg, n_edges);
    dinv_kernel<<<(n_nodes + B - 1) / B, B, 0, stream>>>(deg, dinv, n_nodes);

    // ---- layer 1
    gemm64_scale_wmma<<<gblocks, B, 0, stream>>>(x, W1, dinv, hs, m_tiles);
    fill_kernel<<<fblocks, B, 0, stream>>>(agg, 0.0f, nfeat);
    edge_agg_kernel<<<eblocks, B, 0, stream>>>(hs, src, dst, agg, n_edges);
    finalize_relu_kernel<<<fblocks, B, 0, stream>>>(agg, hs, dinv, b1, hbuf, n_nodes);

    // ---- layer 2
    gemm64_scale_wmma<<<gblocks, B, 0, stream>>>(hbuf, W2, dinv, hs, m_tiles);
    fill_kernel<<<fblocks, B, 0, stream>>>(agg, 0.0f, nfeat);
    edge_agg_kernel<<<eblocks, B, 0, stream>>>(hs, src, dst, agg, n_edges);
    finalize_relu_kernel<<<fblocks, B, 0, stream>>>(agg, hs, dinv, b2, hbuf, n_nodes);

    // ---- pooling + classification
    fill_kernel<<<(n_graphs * (HID + 1) + B - 1) / B, B, 0, stream>>>(pooled, 0.0f, n_graphs * (HID + 1));
    pool_kernel<<<fblocks, B, 0, stream>>>(hbuf, batch, pooled, cnts, n_nodes);
    classify_kernel<<<n_graphs, HID, 0, stream>>>(pooled, cnts, Wc, bc, out);
}